// Method_GCN_Citeseer_22668837388746
// MI455X (gfx1250) — compile-verified
//
#include <hip/hip_runtime.h>
#include <math.h>

typedef __attribute__((ext_vector_type(2))) float v2f;
typedef __attribute__((ext_vector_type(8))) float v8f;

#define F_IN   512
#define HIDDEN 64
#define NCLASS 16
#define SCAN_B 256

// ---------------------------------------------------------------------------
// Generic small transpose: dst[c*K + k] = src[k*C + c]   (src is [K,C])
// ---------------------------------------------------------------------------
__global__ void transpose_kc(const float* __restrict__ src, float* __restrict__ dst,
                             int K, int C) {
    int i = blockIdx.x * blockDim.x + threadIdx.x;
    if (i < K * C) {
        int k = i / C, c = i % C;
        dst[(size_t)c * K + k] = src[i];
    }
}

// ---------------------------------------------------------------------------
// GEMM1: support = x @ W1 via V_WMMA_F32_16X16X4_F32; W1t is [64,512] (transposed)
// One wave -> 16x64 tile; A fragment shared across 4 column tiles per k-step.
// x is streamed once (205 MB): non-temporal loads keep L2 free for SpMM operands.
// ---------------------------------------------------------------------------
__global__ __launch_bounds__(128)
void gemm1_wmma(const float* __restrict__ x, const float* __restrict__ W1t,
                float* __restrict__ support, int N, int ntiles) {
    const int wave = threadIdx.x >> 5;
    const int lane = threadIdx.x & 31;
    const int tile = blockIdx.x * 4 + wave;           // uniform per wave
    if (tile >= ntiles) return;                        // wave-uniform exit, EXEC stays all-1s

    const int row0 = tile * 16;
    const int lo = lane & 15;
    const int hi = lane >> 4;

    const int arow = min(row0 + lo, N - 1);
    const float* __restrict__ xrow = x + (size_t)arow * F_IN;
    const float* __restrict__ w0 = W1t + (size_t)(lo +  0) * F_IN;
    const float* __restrict__ w1 = W1t + (size_t)(lo + 16) * F_IN;
    const float* __restrict__ w2 = W1t + (size_t)(lo + 32) * F_IN;
    const float* __restrict__ w3 = W1t + (size_t)(lo + 48) * F_IN;

    v8f c0 = {}, c1 = {}, c2 = {}, c3 = {};

    for (int k = 0; k < F_IN; k += 4) {
        const int kk = k + 2 * hi;
        v2f a = __builtin_nontemporal_load((const v2f*)(xrow + kk));  // streaming x
        v2f b0 = *(const v2f*)(w0 + kk);
        v2f b1 = *(const v2f*)(w1 + kk);
        v2f b2 = *(const v2f*)(w2 + kk);
        v2f b3 = *(const v2f*)(w3 + kk);
        c0 = __builtin_amdgcn_wmma_f32_16x16x4_f32(false, a, false, b0, (short)0, c0, false, false);
        c1 = __builtin_amdgcn_wmma_f32_16x16x4_f32(false, a, false, b1, (short)0, c1, false, false);
        c2 = __builtin_amdgcn_wmma_f32_16x16x4_f32(false, a, false, b2, (short)0, c2, false, false);
        c3 = __builtin_amdgcn_wmma_f32_16x16x4_f32(false, a, false, b3, (short)0, c3, false, false);
    }

    #pragma unroll
    for (int r = 0; r < 8; ++r) {
        const int row = row0 + r + 8 * hi;
        if (row < N) {
            float* __restrict__ o = support + (size_t)row * HIDDEN + lo;
            o[0]  = c0[r];
            o[16] = c1[r];
            o[32] = c2[r];
            o[48] = c3[r];
        }
    }
}

// ---------------------------------------------------------------------------
// GEMM2: support2 = relu(h) @ W2 ; W2t is [16,64]; ReLU fused into A-load.
// ---------------------------------------------------------------------------
__global__ __launch_bounds__(128)
void gemm2_wmma(const float* __restrict__ h, const float* __restrict__ W2t,
                float* __restrict__ support2, int N, int ntiles) {
    const int wave = threadIdx.x >> 5;
    const int lane = threadIdx.x & 31;
    const int tile = blockIdx.x * 4 + wave;
    if (tile >= ntiles) return;

    const int row0 = tile * 16;
    const int lo = lane & 15;
    const int hi = lane >> 4;

    const int arow = min(row0 + lo, N - 1);
    const float* __restrict__ hrow = h + (size_t)arow * HIDDEN;
    const float* __restrict__ wrow = W2t + (size_t)lo * HIDDEN;

    v8f c = {};
    for (int k = 0; k < HIDDEN; k += 4) {
        const int kk = k + 2 * hi;
        v2f raw = *(const v2f*)(hrow + kk);
        v2f a; a[0] = fmaxf(raw[0], 0.0f); a[1] = fmaxf(raw[1], 0.0f);   // fused ReLU
        v2f b = *(const v2f*)(wrow + kk);
        c = __builtin_amdgcn_wmma_f32_16x16x4_f32(false, a, false, b, (short)0, c, false, false);
    }

    #pragma unroll
    for (int r = 0; r < 8; ++r) {
        const int row = row0 + r + 8 * hi;
        if (row < N) support2[(size_t)row * NCLASS + lo] = c[r];
    }
}

// ---------------------------------------------------------------------------
// CSR-by-dst construction: zero -> histogram -> 2-level exclusive scan -> reorder
// ---------------------------------------------------------------------------
__global__ void zero_i32(int* __restrict__ p, int n) {
    int i = blockIdx.x * blockDim.x + threadIdx.x;
    if (i < n) p[i] = 0;
}

__global__ void hist_dst(const int* __restrict__ edst, int* __restrict__ counts, int E) {
    int i = blockIdx.x * blockDim.x + threadIdx.x;
    if (i < E) atomicAdd(&counts[edst[i]], 1);
}

// per-block exclusive scan (Hillis-Steele in LDS) + block sums
__global__ __launch_bounds__(SCAN_B)
void scan_block(const int* __restrict__ in, int* __restrict__ out,
                int* __restrict__ bsums, int n) {
    __shared__ int tmp[SCAN_B];
    const int t = threadIdx.x;
    const int i = blockIdx.x * SCAN_B + t;
    int v = (i < n) ? in[i] : 0;
    tmp[t] = v;
    __syncthreads();
    #pragma unroll
    for (int off = 1; off < SCAN_B; off <<= 1) {
        int add = (t >= off) ? tmp[t - off] : 0;
        __syncthreads();
        tmp[t] += add;
        __syncthreads();
    }
    if (i < n) out[i] = tmp[t] - v;                    // exclusive
    if (t == SCAN_B - 1) bsums[blockIdx.x] = tmp[t];   // inclusive block total
}

// single-block exclusive scan of block sums (nb <= 512)
__global__ __launch_bounds__(512)
void scan_sums(int* __restrict__ bsums, int nb) {
    __shared__ int tmp[512];
    const int t = threadIdx.x;
    int v = (t < nb) ? bsums[t] : 0;
    tmp[t] = v;
    __syncthreads();
    #pragma unroll
    for (int off = 1; off < 512; off <<= 1) {
        int add = (t >= off) ? tmp[t - off] : 0;
        __syncthreads();
        tmp[t] += add;
        __syncthreads();
    }
    if (t < nb) bsums[t] = tmp[t] - v;                 // exclusive
}

// add scanned block offsets; also seed the reorder cursor
__global__ __launch_bounds__(SCAN_B)
void scan_add(int* __restrict__ row_start, int* __restrict__ cursor,
              const int* __restrict__ bsums, int n) {
    int i = blockIdx.x * SCAN_B + threadIdx.x;
    if (i < n) {
        int s = row_start[i] + bsums[blockIdx.x];
        row_start[i] = s;
        cursor[i] = s;
    }
}

__global__ void reorder_edges(const int* __restrict__ esrc, const int* __restrict__ edst,
                              const float* __restrict__ eval, int* __restrict__ cursor,
                              int* __restrict__ csr_src, float* __restrict__ csr_val, int E) {
    int i = blockIdx.x * blockDim.x + threadIdx.x;
    if (i < E) {
        int d = edst[i];
        int slot = atomicAdd(&cursor[d], 1);
        csr_src[slot] = esrc[i];
        csr_val[slot] = eval[i];
    }
}

// ---------------------------------------------------------------------------
// Gather-side SpMM, D=64: one wave per destination row, bias folded into the
// accumulator, zero atomics. dense gathers are fully coalesced and L2-resident.
// ---------------------------------------------------------------------------
__global__ __launch_bounds__(256)
void spmm_rows64(const int* __restrict__ row_start, const int* __restrict__ counts,
                 const int* __restrict__ csr_src, const float* __restrict__ csr_val,
                 const float* __restrict__ dense, const float* __restrict__ bias,
                 float* __restrict__ out, int N) {
    const int wave = (blockIdx.x * blockDim.x + threadIdx.x) >> 5;
    const int lane = threadIdx.x & 31;
    if (wave >= N) return;                             // wave-uniform
    const int row = wave;

    float a0 = bias[lane];
    float a1 = bias[lane + 32];

    const int s   = __builtin_amdgcn_readfirstlane(row_start[row]);
    const int cnt = __builtin_amdgcn_readfirstlane(counts[row]);
    for (int i = s; i < s + cnt; ++i) {
        const int   src = csr_src[i];                  // uniform -> scalarizable
        const float v   = csr_val[i];
        const float* __restrict__ dp = dense + (size_t)src * HIDDEN;
        a0 = fmaf(v, dp[lane],      a0);
        a1 = fmaf(v, dp[lane + 32], a1);
    }
    out[(size_t)row * HIDDEN + lane]      = a0;
    out[(size_t)row * HIDDEN + lane + 32] = a1;
}

// ---------------------------------------------------------------------------
// Gather-side SpMM, D=16, fused with row softmax. Half-wave owns a row's 16
// logits; softmax via __shfl_xor reductions; writes probabilities directly.
// ---------------------------------------------------------------------------
__global__ __launch_bounds__(256)
void spmm_rows16_softmax(const int* __restrict__ row_start, const int* __restrict__ counts,
                         const int* __restrict__ csr_src, const float* __restrict__ csr_val,
                         const float* __restrict__ dense, const float* __restrict__ bias,
                         float* __restrict__ out, int N) {
    const int gwave = (blockIdx.x * blockDim.x + threadIdx.x) >> 5;
    const int lane = threadIdx.x & 31;
    const int lo = lane & 15;
    const int hi = lane >> 4;
    const int row = gwave * 2 + hi;
    const bool active = row < N;

    float acc = 0.0f;
    int s = 0, cnt = 0;
    if (active) {
        acc = bias[lo];
        s   = row_start[row];
        cnt = counts[row];
    }
    for (int i = s; i < s + cnt; ++i) {
        const int   src = csr_src[i];
        const float v   = csr_val[i];
        acc = fmaf(v, dense[(size_t)src * NCLASS + lo], acc);
    }

    // softmax over 16 lanes of the half-wave
    float m = acc;
    #pragma unroll
    for (int off = 8; off >= 1; off >>= 1) m = fmaxf(m, __shfl_xor(m, off, 16));
    const float ex = __expf(acc - m);
    float sum = ex;
    #pragma unroll
    for (int off = 8; off >= 1; off >>= 1) sum += __shfl_xor(sum, off, 16);

    if (active) out[(size_t)row * NCLASS + lo] = ex / sum;
}

// ---------------------------------------------------------------------------
// Launcher
// ---------------------------------------------------------------------------
extern "C" void kernel_launch(void* const* d_in, const int* in_sizes, int n_in,
                              void* d_out, int out_size, void* d_ws, size_t ws_size,
                              hipStream_t stream) {
    const float* x    = (const float*)d_in[0];
    const int*   esrc = (const int*)  d_in[1];
    const int*   edst = (const int*)  d_in[2];
    const float* eval = (const float*)d_in[3];
    const float* W1   = (const float*)d_in[4];
    const float* b1   = (const float*)d_in[5];
    const float* W2   = (const float*)d_in[6];
    const float* b2   = (const float*)d_in[7];
    float* out = (float*)d_out;

    const int N = in_sizes[0] / F_IN;     // 100000
    const int E = in_sizes[1];            // 1600000

    // ---- workspace carve-up (~72 MB) ----
    float* support1 = (float*)d_ws;                       // N*64
    float* hbuf     = support1 + (size_t)N * HIDDEN;      // N*64
    float* support2 = hbuf     + (size_t)N * HIDDEN;      // N*16
    float* W1t      = support2 + (size_t)N * NCLASS;      // 64*512
    float* W2t      = W1t + F_IN * HIDDEN;                // 16*64
    float* csr_val  = W2t + HIDDEN * NCLASS;              // E
    int*   counts   = (int*)(csr_val + (size_t)E);        // N
    int*   rstart   = counts + N;                         // N
    int*   cursor   = rstart + N;                         // N
    int*   bsums    = cursor + N;                         // <=1024
    int*   csr_src  = bsums + 1024;                       // E

    const int ntiles = (N + 15) / 16;                     // 6250
    const dim3 gemm_grid((ntiles + 3) / 4);
    const int nscan = (N + SCAN_B - 1) / SCAN_B;          // 391 (<512)

    // weight transposes (tiny) so WMMA B-fragments are contiguous v2f loads
    transpose_kc<<<(F_IN * HIDDEN + 255) / 256, 256, 0, stream>>>(W1, W1t, F_IN, HIDDEN);
    transpose_kc<<<(HIDDEN * NCLASS + 255) / 256, 256, 0, stream>>>(W2, W2t, HIDDEN, NCLASS);

    // CSR-by-dst build: histogram -> scan -> reorder (one-time, reused by both layers)
    zero_i32<<<(N + 255) / 256, 256, 0, stream>>>(counts, N);
    hist_dst<<<(E + 255) / 256, 256, 0, stream>>>(edst, counts, E);
    scan_block<<<nscan, SCAN_B, 0, stream>>>(counts, rstart, bsums, N);
    scan_sums<<<1, 512, 0, stream>>>(bsums, nscan);
    scan_add<<<nscan, SCAN_B, 0, stream>>>(rstart, cursor, bsums, N);
    reorder_edges<<<(E + 255) / 256, 256, 0, stream>>>(esrc, edst, eval, cursor,
                                                       csr_src, csr_val, E);

    // Layer 1: support1 = x @ W1 ; h = A @ support1 + b1 (bias folded, no atomics)
    gemm1_wmma<<<gemm_grid, 128, 0, stream>>>(x, W1t, support1, N, ntiles);
    {
        long long threads = (long long)N * 32;            // one wave per row
        spmm_rows64<<<(unsigned)((threads + 255) / 256), 256, 0, stream>>>(
            rstart, counts, csr_src, csr_val, support1, b1, hbuf, N);
    }

    // Layer 2: support2 = relu(h) @ W2 ; out = softmax(A @ support2 + b2) fused
    gemm2_wmma<<<gemm_grid, 128, 0, stream>>>(hbuf, W2t, support2, N, ntiles);
    {
        int nwaves = (N + 1) / 2;                         // half-wave per row
        long long threads = (long long)nwaves * 32;
        spmm_rows16_softmax<<<(unsigned)((threads + 255) / 256), 256, 0, stream>>>(
            rstart, counts, csr_src, csr_val, support2, b2, out, N);
    }
}